// OuterProduct_23476291239988
// MI455X (gfx1250) — compile-verified
//
#include <hip/hip_runtime.h>

#define B_   2
#define N_   768
#define CIN  384
#define COUT 32
#define DOUT 128
#define ROWS (B_ * N_)   // 1536
#define EPSF 1e-5f

typedef __attribute__((ext_vector_type(2))) float v2f;
typedef __attribute__((ext_vector_type(8))) float v8f;

// ---------------------------------------------------------------------------
// Kernel 0: pre-sum projection weights over their output dim.
//   wl_sum[c] = sum_o Wl[o,c], wr_sum[c] = sum_o Wr[o,c], plus bias sums.
// One block of CIN threads; trivial.
// ---------------------------------------------------------------------------
__global__ void prep_weights(const float* __restrict__ Wl, const float* __restrict__ bl,
                             const float* __restrict__ Wr, const float* __restrict__ br,
                             float* __restrict__ ws) {
    int c = threadIdx.x;                    // 0..383
    float sl = 0.f, sr = 0.f;
    for (int o = 0; o < COUT; ++o) {
        sl += Wl[o * CIN + c];
        sr += Wr[o * CIN + c];
    }
    ws[c]       = sl;
    ws[CIN + c] = sr;
    if (c == 0) { float s = 0.f; for (int o = 0; o < COUT; ++o) s += bl[o]; ws[2 * CIN]     = s; }
    if (c == 1) { float s = 0.f; for (int o = 0; o < COUT; ++o) s += br[o]; ws[2 * CIN + 1] = s; }
}

// ---------------------------------------------------------------------------
// Kernel 1: per-row LayerNorm + two dots with the pre-summed weights.
//   ls[r] = normed[r,:] . wl_sum + sum(bl);  rs[r] likewise.
// One block (128 threads) per row; each thread owns 3 channels.
// ---------------------------------------------------------------------------
__global__ __launch_bounds__(128) void ln_rowdots(const float* __restrict__ act,
                                                  const float* __restrict__ gamma,
                                                  const float* __restrict__ beta,
                                                  const float* __restrict__ ws,
                                                  float* __restrict__ lsp,
                                                  float* __restrict__ rsp) {
    __shared__ float redA[128];
    __shared__ float redB[128];
    const int r = blockIdx.x;
    const int t = threadIdx.x;
    const float* x = act + (size_t)r * CIN;

    float xv[3];
    float s = 0.f, sq = 0.f;
    #pragma unroll
    for (int k = 0; k < 3; ++k) {
        float v = x[t + 128 * k];
        xv[k] = v; s += v; sq += v * v;
    }
    redA[t] = s; redB[t] = sq;
    __syncthreads();
    for (int off = 64; off > 0; off >>= 1) {
        if (t < off) { redA[t] += redA[t + off]; redB[t] += redB[t + off]; }
        __syncthreads();
    }
    const float mu   = redA[0] * (1.0f / CIN);
    const float var  = redB[0] * (1.0f / CIN) - mu * mu;
    const float rstd = rsqrtf(var + EPSF);
    __syncthreads();

    float dl = 0.f, dr = 0.f;
    #pragma unroll
    for (int k = 0; k < 3; ++k) {
        int c = t + 128 * k;
        float nv = (xv[k] - mu) * rstd * gamma[c] + beta[c];
        dl += nv * ws[c];
        dr += nv * ws[CIN + c];
    }
    redA[t] = dl; redB[t] = dr;
    __syncthreads();
    for (int off = 64; off > 0; off >>= 1) {
        if (t < off) { redA[t] += redA[t + off]; redB[t] += redB[t + off]; }
        __syncthreads();
    }
    if (t == 0) {
        lsp[r] = redA[0] + ws[2 * CIN];
        rsp[r] = redB[0] + ws[2 * CIN + 1];
    }
}

// ---------------------------------------------------------------------------
// Kernel 2 (hot, store-bandwidth-bound): expand
//   out[b,i,j,d] = ls[b,i]*rs[b,j]*w_act[d] + b_act[d]
// via exact-fp32 V_WMMA_F32_16X16X4_F32 rank-1 tiles:
//   A[m,0] = ls_i*rs[j0+m] (K padded with zeros), B[0,n] = w_act[d0+n],
//   C[m,n] = b_act[d0+n]  =>  D = A*B + C is the 16x16 (j x d) output tile.
// Block = 256 threads = 8 waves; wave w owns d-tile w (8 tiles of 16 = 128).
// ---------------------------------------------------------------------------
__global__ __launch_bounds__(256) void outer_wmma(const float* __restrict__ lsp,
                                                  const float* __restrict__ rsp,
                                                  const float* __restrict__ w_act,
                                                  const float* __restrict__ b_act,
                                                  float* __restrict__ out) {
    const int jt   = blockIdx.x % (N_ / 16);     // 0..47
    const int r    = blockIdx.x / (N_ / 16);     // 0..1535 == b*N_ + i
    const int bidx = r / N_;
    const int wave = threadIdx.x >> 5;           // 0..7 -> d tile
    const int lane = threadIdx.x & 31;
    const int lo   = lane & 15;
    const int d0   = wave * 16;
    const int j0   = jt * 16;
    const bool lowHalf = (lane < 16);

    const float ls_i = lsp[r];

    // A-matrix (16x4 f32): lanes 0-15 hold {K=0,K=1}, lanes 16-31 hold {K=2,K=3}.
    // Only K=0 column is non-zero (rank-1 update).
    v2f a;
    a.x = lowHalf ? ls_i * rsp[bidx * N_ + j0 + lo] : 0.0f;
    a.y = 0.0f;

    // B-matrix (4x16 f32): row K=0 = w_act tile, rows 1-3 zero.
    v2f bm;
    bm.x = lowHalf ? w_act[d0 + lo] : 0.0f;
    bm.y = 0.0f;

    // C-matrix: C[m,n] = b_act[d0+n] for every m -> same value in all 8 VGPRs.
    const float bb = b_act[d0 + lo];
    v8f c = {bb, bb, bb, bb, bb, bb, bb, bb};

    v8f d = __builtin_amdgcn_wmma_f32_16x16x4_f32(
        /*neg_a=*/false, a, /*neg_b=*/false, bm,
        /*c_mod=*/(short)0, c, /*reuse_a=*/false, /*reuse_b=*/false);

    // D layout: VGPR g, lane l -> M = g + 8*(l>=16) (= j offset), N = l&15 (= d offset).
    const int jbase = j0 + ((lane >> 4) << 3);
    float* op = out + ((size_t)r * N_ + jbase) * DOUT + d0 + lo;
    #pragma unroll
    for (int g = 0; g < 8; ++g) {
        op[(size_t)g * DOUT] = d[g];
    }
}

// ---------------------------------------------------------------------------
extern "C" void kernel_launch(void* const* d_in, const int* in_sizes, int n_in,
                              void* d_out, int out_size, void* d_ws, size_t ws_size,
                              hipStream_t stream) {
    const float* act   = (const float*)d_in[0];
    const float* gamma = (const float*)d_in[1];
    const float* beta  = (const float*)d_in[2];
    const float* Wl    = (const float*)d_in[3];
    const float* bl    = (const float*)d_in[4];
    const float* Wr    = (const float*)d_in[5];
    const float* br    = (const float*)d_in[6];
    const float* w_act = (const float*)d_in[7];
    const float* b_act = (const float*)d_in[8];
    float* out = (float*)d_out;

    // Workspace layout (floats): [0,384) wl_sum | [384,768) wr_sum |
    // [768,770) bias sums | [800,2336) ls | [2336,3872) rs
    float* ws  = (float*)d_ws;
    float* lsp = ws + 800;
    float* rsp = ws + 800 + ROWS;

    prep_weights<<<1, CIN, 0, stream>>>(Wl, bl, Wr, br, ws);
    ln_rowdots<<<ROWS, 128, 0, stream>>>(act, gamma, beta, ws, lsp, rsp);
    outer_wmma<<<ROWS * (N_ / 16), 256, 0, stream>>>(lsp, rsp, w_act, b_act, out);
}